// CRF_39264591020355
// MI455X (gfx1250) — compile-verified
//
#include <hip/hip_runtime.h>
#include <hip/hip_bf16.h>
#include <math.h>

// Problem constants (from reference setup_inputs): B=512, S=512, C=96
#define Bc 512
#define Sc 512
#define Cc 96
#define PSTR 100          // padded LDS row stride for p-matrix (kills bank conflicts)
#define KC 24             // K chunks of 4 (96 / 4) for V_WMMA_F32_16X16X4_F32
#define NWAVE 6           // 6 waves per WG, one 16-wide j-tile each (6*16 = 96)
#define RENORM 8          // exact renormalization period (drift-safe, see analysis)

typedef float v2f __attribute__((ext_vector_type(2)));
typedef float v8f __attribute__((ext_vector_type(8)));

#if defined(__has_builtin)
#if __has_builtin(__builtin_amdgcn_wmma_f32_16x16x4_f32)
#define USE_WMMA_F32 1
#endif
#endif

// ---------------------------------------------------------------------------
// Kernel 1: expT = exp(transitions), 96x96 fp32
// ---------------------------------------------------------------------------
__global__ void crf_expT_kernel(const float* __restrict__ T, float* __restrict__ expT) {
    int i = blockIdx.x * blockDim.x + threadIdx.x;
    if (i < Cc * Cc) expT[i] = __expf(T[i]);
}

// ---------------------------------------------------------------------------
// Kernel 2: numerator per batch row (gather score along the gold path)
// ---------------------------------------------------------------------------
__global__ void crf_numerator_kernel(const float* __restrict__ em,
                                     const int* __restrict__ tags,
                                     const unsigned char* __restrict__ masks,
                                     const float* __restrict__ startT,
                                     const float* __restrict__ endT,
                                     const float* __restrict__ T,
                                     float* __restrict__ num) {
    int b = blockIdx.x * blockDim.x + threadIdx.x;
    if (b >= Bc) return;
    const size_t base = (size_t)b * Sc;
    int prev = tags[base];
    float sc = startT[prev] + em[base * Cc + prev];
    int cnt = masks[base] ? 1 : 0;
    for (int s = 1; s < Sc; ++s) {
        int tg = tags[base + s];
        if (masks[base + s]) {
            sc += T[prev * Cc + tg] + em[(base + s) * Cc + tg];
            cnt++;
        }
        prev = tg;  // reference uses raw consecutive tags
    }
    int lastTag = tags[base + (cnt - 1)];
    num[b] = sc + endT[lastTag];
}

// ---------------------------------------------------------------------------
// Kernel 3: forward scan via FP32 WMMA, lazy renormalization.
//  alpha[b,j] = off[b] + s*D + log(P[b,j]);  cheap step:
//      P' = (P · expT) * exp(em - D)         (pure linear algebra + 8 exps)
//  exact renorm (row-max rescale) every RENORM steps. One barrier per cheap
//  step (double-buffered P in LDS); two WMMA accumulator chains.
// ---------------------------------------------------------------------------
__global__ __launch_bounds__(192)
void crf_forward_kernel(const float* __restrict__ em,
                        const unsigned char* __restrict__ masks,
                        const float* __restrict__ startT,
                        const float* __restrict__ endT,
                        const float* __restrict__ expTg,
                        float* __restrict__ den) {
    __shared__ float        sP[2][16 * PSTR];        // double-buffered p matrix
    __shared__ unsigned int sMaskW[16 * (Sc / 4)];   // mask tile, 8KB
    __shared__ float        sPartial[16 * 8];        // [row][wave] partial maxes
    __shared__ float        sRowInv[16];             // 1/rowmax (renorm) or rowmax (init)
    __shared__ float        sOff[16];                // offsets (wave0 owns)
    __shared__ float        sDenPart[16 * 8];

    const int tid  = threadIdx.x;
    const int wave = tid >> 5;        // 0..5
    const int lane = tid & 31;
    const int ln   = lane & 15;       // N (column within tile) / A-row M
    const int hi   = lane >> 4;       // half-wave selector
    const int jb   = wave * 16;       // this wave's j-tile base
    const int rowBase = blockIdx.x * 16;

    const float Dconst = 5.0f;
    const float eNegD  = 0.006737946999085467f;   // exp(-5)

    const unsigned char* smb = reinterpret_cast<const unsigned char*>(sMaskW);

    // ---- preload mask tile into LDS (16 rows x 512 bytes) ----
    for (int idx = tid; idx < 16 * (Sc / 4); idx += 192) {
        int r = idx / (Sc / 4), w = idx % (Sc / 4);
        sMaskW[r * (Sc / 4) + w] =
            reinterpret_cast<const unsigned int*>(masks)[((size_t)(rowBase + r) * Sc) / 4 + w];
    }

    // ---- preload expT B-fragments into registers (reused for all steps) ----
    v2f Bfrag[KC];
#pragma unroll
    for (int k = 0; k < KC; ++k) {
        int r0 = k * 4 + 2 * hi;
        Bfrag[k].x = expTg[r0 * Cc + jb + ln];
        Bfrag[k].y = expTg[(r0 + 1) * Cc + jb + ln];
    }
    const float eendj = __expf(endT[jb + ln]);

    if (wave == 0 && lane < 16) sOff[lane] = 0.0f;

    // ---- init: alpha0 = start + emissions[:,0,:]; exact normalize -> sP[1] ----
    float pP[8];
    {
        float q[8];
        const float st = startT[jb + ln];
#pragma unroll
        for (int r = 0; r < 8; ++r) {
            int M = r + 8 * hi;
            q[r] = st + em[((size_t)(rowBase + M) * Sc) * Cc + jb + ln];
        }
        float mx[8];
#pragma unroll
        for (int r = 0; r < 8; ++r) {
            float m = q[r];
#pragma unroll
            for (int off = 8; off > 0; off >>= 1)
                m = fmaxf(m, __shfl_xor(m, off, 16));
            mx[r] = m;
        }
        if (ln == 0) {
#pragma unroll
            for (int r = 0; r < 8; ++r) sPartial[(r + 8 * hi) * 8 + wave] = mx[r];
        }
        __syncthreads();
        if (wave == 0 && lane < 16) {
            float f = -3.0e38f;
#pragma unroll
            for (int w = 0; w < NWAVE; ++w) f = fmaxf(f, sPartial[lane * 8 + w]);
            sRowInv[lane] = f;       // holds rowmax for init
            sOff[lane] += f;
        }
        __syncthreads();
#pragma unroll
        for (int r = 0; r < 8; ++r) {
            int M = r + 8 * hi;
            float p = __expf(q[r] - sRowInv[M]);
            pP[r] = p;
            sP[1][M * PSTR + jb + ln] = p;
        }
        __syncthreads();
    }

    // ---- main scan over s = 1..S-1 ----
    float ecur[8];
#pragma unroll
    for (int r = 0; r < 8; ++r)
        ecur[r] = em[((size_t)(rowBase + r + 8 * hi) * Sc + 1) * Cc + jb + ln];

    for (int s = 1; s < Sc; ++s) {
        const int rb = s & 1;            // read buffer
        const int wb = rb ^ 1;           // write buffer

        // prefetch next step's emissions
        float enext[8] = {0, 0, 0, 0, 0, 0, 0, 0};
        if (s + 1 < Sc) {
#pragma unroll
            for (int r = 0; r < 8; ++r)
                enext[r] = em[((size_t)(rowBase + r + 8 * hi) * Sc + (s + 1)) * Cc + jb + ln];
        }

        // acc[b,j] = sum_i P[b,i] * expT[i,j] -- two accumulator chains
        const float* sPr = &sP[rb][0];
        v8f acc0 = {0.f, 0.f, 0.f, 0.f, 0.f, 0.f, 0.f, 0.f};
        v8f acc1 = {0.f, 0.f, 0.f, 0.f, 0.f, 0.f, 0.f, 0.f};
#ifdef USE_WMMA_F32
#pragma unroll
        for (int k = 0; k < KC; k += 2) {
            float2 av0 = *reinterpret_cast<const float2*>(&sPr[ln * PSTR + (k * 4 + 2 * hi)]);
            float2 av1 = *reinterpret_cast<const float2*>(&sPr[ln * PSTR + ((k + 1) * 4 + 2 * hi)]);
            v2f a0; a0.x = av0.x; a0.y = av0.y;
            v2f a1; a1.x = av1.x; a1.y = av1.y;
            acc0 = __builtin_amdgcn_wmma_f32_16x16x4_f32(
                false, a0, false, Bfrag[k], (short)0, acc0, false, false);
            acc1 = __builtin_amdgcn_wmma_f32_16x16x4_f32(
                false, a1, false, Bfrag[k + 1], (short)0, acc1, false, false);
        }
#else
        for (int i = 0; i < Cc; ++i) {
            float w = expTg[i * Cc + jb + ln];
#pragma unroll
            for (int r = 0; r < 8; ++r) acc0[r] += sPr[(r + 8 * hi) * PSTR + i] * w;
        }
#endif
        v8f accv = acc0 + acc1;

        // cheap update: P' = mask ? acc * exp(em - D) : P * exp(-D)
        float v[8];
#pragma unroll
        for (int r = 0; r < 8; ++r) {
            int M = r + 8 * hi;
            unsigned char mk = smb[M * Sc + s];
            v[r] = mk ? accv[r] * __expf(ecur[r] - Dconst) : pP[r] * eNegD;
        }

        if ((s & (RENORM - 1)) == 0) {
            // exact renormalization: divide each row by its max, fold into offset
            float mx[8];
#pragma unroll
            for (int r = 0; r < 8; ++r) {
                float m = v[r];
#pragma unroll
                for (int off = 8; off > 0; off >>= 1)
                    m = fmaxf(m, __shfl_xor(m, off, 16));
                mx[r] = m;
            }
            if (ln == 0) {
#pragma unroll
                for (int r = 0; r < 8; ++r) sPartial[(r + 8 * hi) * 8 + wave] = mx[r];
            }
            __syncthreads();
            if (wave == 0 && lane < 16) {
                float f = 0.0f;
#pragma unroll
                for (int w = 0; w < NWAVE; ++w) f = fmaxf(f, sPartial[lane * 8 + w]);
                sRowInv[lane] = 1.0f / f;
                sOff[lane] += __logf(f);
            }
            __syncthreads();
#pragma unroll
            for (int r = 0; r < 8; ++r) v[r] *= sRowInv[r + 8 * hi];
        }

        // store P' and advance
#pragma unroll
        for (int r = 0; r < 8; ++r) {
            int M = r + 8 * hi;
            pP[r] = v[r];
            sP[wb][M * PSTR + jb + ln] = v[r];
            ecur[r] = enext[r];
        }
        __syncthreads();
    }

    // ---- denominator: off + (S-1)*D + log(sum_j P[b,j] * exp(end[j])) ----
    {
#pragma unroll
        for (int r = 0; r < 8; ++r) {
            float t = pP[r] * eendj;
#pragma unroll
            for (int off = 8; off > 0; off >>= 1)
                t += __shfl_xor(t, off, 16);
            if (ln == 0) sDenPart[(r + 8 * hi) * 8 + wave] = t;
        }
        __syncthreads();
        if (wave == 0 && lane < 16) {
            float ssum = 0.f;
#pragma unroll
            for (int w = 0; w < NWAVE; ++w) ssum += sDenPart[lane * 8 + w];
            den[rowBase + lane] = sOff[lane] + Dconst * (float)(Sc - 1) + __logf(ssum);
        }
    }
}

// ---------------------------------------------------------------------------
// Kernel 4: out = mean(num - den)
// ---------------------------------------------------------------------------
__global__ void crf_reduce_kernel(const float* __restrict__ num,
                                  const float* __restrict__ den,
                                  float* __restrict__ out) {
    __shared__ float red[256];
    int tid = threadIdx.x;
    float v = 0.f;
    for (int i = tid; i < Bc; i += 256) v += num[i] - den[i];
    red[tid] = v;
    __syncthreads();
    for (int sft = 128; sft > 0; sft >>= 1) {
        if (tid < sft) red[tid] += red[tid + sft];
        __syncthreads();
    }
    if (tid == 0) out[0] = red[0] / (float)Bc;
}

// ---------------------------------------------------------------------------
extern "C" void kernel_launch(void* const* d_in, const int* in_sizes, int n_in,
                              void* d_out, int out_size, void* d_ws, size_t ws_size,
                              hipStream_t stream) {
    const float*         em     = (const float*)d_in[0];
    const int*           tags   = (const int*)d_in[1];
    const unsigned char* masks  = (const unsigned char*)d_in[2];
    const float*         startT = (const float*)d_in[3];
    const float*         endT   = (const float*)d_in[4];
    const float*         T      = (const float*)d_in[5];
    float*               out    = (float*)d_out;

    float* expT = (float*)d_ws;          // 96*96 floats
    float* num  = expT + Cc * Cc;        // 512 floats
    float* den  = num + Bc;              // 512 floats

    crf_expT_kernel<<<dim3((Cc * Cc + 255) / 256), dim3(256), 0, stream>>>(T, expT);
    crf_numerator_kernel<<<dim3((Bc + 255) / 256), dim3(256), 0, stream>>>(
        em, tags, masks, startT, endT, T, num);
    crf_forward_kernel<<<dim3(Bc / 16), dim3(192), 0, stream>>>(
        em, masks, startT, endT, expT, den);
    crf_reduce_kernel<<<dim3(1), dim3(256), 0, stream>>>(num, den, out);
}